// SFFM_module_57724360458679
// MI455X (gfx1250) — compile-verified
//
#include <hip/hip_runtime.h>
#include <hip/hip_bf16.h>

// SFFM: gap = mean(inputs, HW); scores = gap @ W^T; attn = softmax_L(scores);
// out = inputs * attn.  L=4, B=8, C=256, H=W=64.
//
// Memory-bound (256 MB HBM traffic min). Strategy for MI455X:
//  - K1 (GAP) reads input with RT hints -> 128 MB lands in the 192 MB L2.
//  - K2 (GEMM) uses V_WMMA_F32_16X16X4_F32 (exact fp32 WMMA).
//  - K3 (apply) re-reads input non-temporally (L2 hits, last use) and writes
//    the output with non-temporal stores so it does not evict the input.

typedef float v2f __attribute__((ext_vector_type(2)));
typedef float v4f __attribute__((ext_vector_type(4)));
typedef float v8f __attribute__((ext_vector_type(8)));

#define Ldim 4
#define Bdim 8
#define Cdim 256
#define HW   4096   // 64*64

// ---------------------------------------------------------------- K1: GAP ---
// One 256-thread block per (l,b,c) row; 4096 contiguous floats -> mean.
__global__ void __launch_bounds__(256) sffm_gap_kernel(
    const float* __restrict__ in, float* __restrict__ gap) {
  __shared__ float sdata[256];
  const int row = blockIdx.x;                 // (l*B + b)*C + c, 0..8191
  const int t   = threadIdx.x;
  const v4f* p = (const v4f*)(in + (size_t)row * HW);
  float s = 0.0f;
#pragma unroll
  for (int i = 0; i < 4; ++i) {               // 1024 float4 / 256 threads
    v4f v = p[t + i * 256];
    s += (v.x + v.y) + (v.z + v.w);
  }
  sdata[t] = s;
  __syncthreads();
  for (int off = 128; off > 0; off >>= 1) {
    if (t < off) sdata[t] += sdata[t + off];
    __syncthreads();
  }
  if (t == 0) gap[row] = sdata[0] * (1.0f / (float)HW);
}

// --------------------------------------------------- K2: scores via WMMA ---
// scores[m][n] = sum_k gap[m][k] * W[n][k],  m = l*8+b (32 rows), n,k in 256.
// One wave (32 threads) per 16x16 D tile; 64 steps of V_WMMA_F32_16X16X4_F32.
// A (16x4 fp32) layout: lanes 0-15 -> M=lane, v0=K0 v1=K1; lanes 16-31 -> K2,K3.
// B (4x16) mirrors: v0 = row K(+half*2), v1 = row K+1, N = lane%16.
__global__ void __launch_bounds__(32) sffm_gemm_kernel(
    const float* __restrict__ gap, const float* __restrict__ Wlin,
    float* __restrict__ scores) {
  const int lane  = threadIdx.x;              // 0..31 (full wave, EXEC all-1s)
  const int mtile = blockIdx.x >> 4;          // 0..1
  const int ntile = blockIdx.x & 15;          // 0..15
  const int half  = lane >> 4;                // 0 or 1
  const int lr    = lane & 15;
  const int m = mtile * 16 + lr;              // A row for this lane
  const int n = ntile * 16 + lr;              // B col for this lane

  v8f acc = {};
  for (int kk = 0; kk < Cdim; kk += 4) {
    const int k = kk + half * 2;
    v2f a, b;
    a.x = gap[m * Cdim + k];
    a.y = gap[m * Cdim + k + 1];
    b.x = Wlin[n * Cdim + k];                 // B[k][n] = W[n][k]
    b.y = Wlin[n * Cdim + k + 1];
    acc = __builtin_amdgcn_wmma_f32_16x16x4_f32(
        /*neg_a=*/false, a, /*neg_b=*/false, b,
        /*c_mod=*/(short)0, acc, /*reuse_a=*/false, /*reuse_b=*/false);
  }

  const int col = ntile * 16 + lr;
#pragma unroll
  for (int v = 0; v < 8; ++v) {               // D layout: VGPR v -> row v(+8)
    const int row = mtile * 16 + v + half * 8;
    scores[row * Cdim + col] = acc[v];
  }
}

// --------------------------------------- K3: softmax over L + broadcast mul --
// One 256-thread block per (b,c); softmax of 4 uniform scalars, then stream
// 4 levels x 4096 floats with NT load (last use) / NT store (write-once).
__global__ void __launch_bounds__(256) sffm_apply_kernel(
    const float* __restrict__ in, const float* __restrict__ scores,
    float* __restrict__ out) {
  const int bc = blockIdx.x;                  // b*C + c, 0..2047
  const int b  = bc >> 8;
  const int c  = bc & 255;
  const int t  = threadIdx.x;

  float s[Ldim];
#pragma unroll
  for (int l = 0; l < Ldim; ++l)
    s[l] = scores[((l << 3) + b) * Cdim + c];

  float mx = fmaxf(fmaxf(s[0], s[1]), fmaxf(s[2], s[3]));
  float e[Ldim];
  float sum = 0.0f;
#pragma unroll
  for (int l = 0; l < Ldim; ++l) { e[l] = __expf(s[l] - mx); sum += e[l]; }
  const float inv = 1.0f / sum;

#pragma unroll
  for (int l = 0; l < Ldim; ++l) {
    const float a = e[l] * inv;
    const size_t base = (size_t)(((l << 3) + b) * Cdim + c) * HW;
    const v4f* ip = (const v4f*)(in + base);
    v4f*       op = (v4f*)(out + base);
#pragma unroll
    for (int i = 0; i < 4; ++i) {
      v4f v = __builtin_nontemporal_load(&ip[t + i * 256]);
      v4f r = v * a;
      __builtin_nontemporal_store(r, &op[t + i * 256]);
    }
  }
}

// ---------------------------------------------------------------------------
extern "C" void kernel_launch(void* const* d_in, const int* in_sizes, int n_in,
                              void* d_out, int out_size, void* d_ws, size_t ws_size,
                              hipStream_t stream) {
  const float* in   = (const float*)d_in[0];   // [4,8,256,64,64] fp32
  const float* Wlin = (const float*)d_in[1];   // [256,256] fp32
  float* out    = (float*)d_out;               // [4,8,256,64,64] fp32
  float* gap    = (float*)d_ws;                // 8192 floats
  float* scores = gap + Ldim * Bdim * Cdim;    // 8192 floats

  sffm_gap_kernel  <<<Ldim * Bdim * Cdim, 256, 0, stream>>>(in, gap);
  sffm_gemm_kernel <<<32, 32, 0, stream>>>(gap, Wlin, scores);
  sffm_apply_kernel<<<Bdim * Cdim, 256, 0, stream>>>(in, scores, out);
}